// GCNRFPEncode_33552284516502
// MI455X (gfx1250) — compile-verified
//
#include <hip/hip_runtime.h>

typedef float v2f __attribute__((ext_vector_type(2)));
typedef float v8f __attribute__((ext_vector_type(8)));

#define BN_EPS 1e-3f

// ---------------------------------------------------------------------------
// Kernel 1: degree histogram  deg[src] += 1  (one thread per edge)
// ---------------------------------------------------------------------------
__global__ __launch_bounds__(256) void deg_kernel(const int* __restrict__ ei,
                                                  float* __restrict__ deg, int E) {
  int e = blockIdx.x * blockDim.x + threadIdx.x;
  if (e < E) atomicAdd(&deg[ei[2 * e]], 1.0f);
}

// ---------------------------------------------------------------------------
// Kernel 2: inv_sqrt_deg[i] = deg[i]^-0.5
// ---------------------------------------------------------------------------
__global__ __launch_bounds__(256) void invd_kernel(const float* __restrict__ deg,
                                                   float* __restrict__ invd, int n) {
  int i = blockIdx.x * blockDim.x + threadIdx.x;
  if (i < n) invd[i] = rsqrtf(deg[i]);
}

// ---------------------------------------------------------------------------
// Kernel 3: h = X (N x 128) @ W (128 x 64) + b, via V_WMMA_F32_16X16X4_F32.
// One wave computes one 16x16 tile of h. 8 waves (256 threads) per block.
// K loop: 32 chained wmma_f32_16x16x4_f32 ops.
//
// VGPR layouts (CDNA5 ISA 7.12.2):
//   A (16x4, MxK):  lane l, vgpr v -> A[l%16][2*(l>>4)+v]   (contiguous float2)
//   B (4x16, KxN):  lane l, vgpr v -> B[2*(l>>4)+v][l%16]
//   C/D (16x16):    lane l, vgpr v -> C[v + 8*(l>>4)][l%16]
// ---------------------------------------------------------------------------
__global__ __launch_bounds__(256) void gemm_wmma(const float* __restrict__ X,
                                                 const float* __restrict__ W,
                                                 const float* __restrict__ bias,
                                                 float* __restrict__ Hout, int Nrows) {
  const int F = 128, Hc = 64;
  const int lane = threadIdx.x & 31;
  const int wave = threadIdx.x >> 5;
  const int tile = blockIdx.x * 8 + wave;
  const int ntiles = ((Nrows + 15) >> 4) * 4;   // 4 column tiles of 16 (H=64)
  if (tile >= ntiles) return;                   // wave-uniform: EXEC stays all-1s

  const int row0 = (tile >> 2) << 4;
  const int col0 = (tile & 3) << 4;
  const int m  = lane & 15;                     // position within 16
  const int hi = lane >> 4;                     // half-wave: 0 or 1

  const float* __restrict__ xrow = X + (size_t)(row0 + m) * F;
  const float* __restrict__ wcol = W + col0 + m;

  v8f acc = {};
#pragma unroll
  for (int k = 0; k < F; k += 4) {
    const int ka = k + 2 * hi;
    v2f a = *reinterpret_cast<const v2f*>(xrow + ka);     // A[m][ka], A[m][ka+1]
    v2f b;
    b.x = wcol[(size_t)ka * Hc];                          // W[ka]  [col0+m]
    b.y = wcol[(size_t)(ka + 1) * Hc];                    // W[ka+1][col0+m]
    acc = __builtin_amdgcn_wmma_f32_16x16x4_f32(
        /*neg_a=*/false, a, /*neg_b=*/false, b,
        /*c_mod=*/(short)0, acc, /*reuse_a=*/false, /*reuse_b=*/false);
  }

  const float bv = bias[col0 + m];
#pragma unroll
  for (int v = 0; v < 8; ++v) {
    const int row = row0 + v + 8 * hi;
    Hout[(size_t)row * Hc + (col0 + m)] = acc[v] + bv;
  }
}

// ---------------------------------------------------------------------------
// Kernel 4: edge scatter. agg[src] += 0.5 * invd[src]*invd[dst] * h[dst].
// 32 threads (one wave) per edge; each lane handles 2 consecutive columns
// (float2 coalesced read of h[dst], two f32 atomics into agg[src]).
// h and agg (25.6 MB each) live in the 192 MB L2, so the random gather and
// the atomic scatter both resolve on-chip.
// ---------------------------------------------------------------------------
__global__ __launch_bounds__(256) void edge_scatter(const int* __restrict__ ei,
                                                    const float* __restrict__ h,
                                                    const float* __restrict__ invd,
                                                    float* __restrict__ agg, int E) {
  long long gid = (long long)blockIdx.x * blockDim.x + threadIdx.x;
  int e = (int)(gid >> 5);
  if (e >= E) return;
  int j = ((int)gid & 31) << 1;                 // column pair 0..62
  int s = ei[2 * e];
  int d = ei[2 * e + 1];
  float scale = 0.5f * invd[s] * invd[d];
  const float2 hv = *reinterpret_cast<const float2*>(h + (size_t)d * 64 + j);
  atomicAdd(agg + (size_t)s * 64 + j,     scale * hv.x);
  atomicAdd(agg + (size_t)s * 64 + j + 1, scale * hv.y);
}

// ---------------------------------------------------------------------------
// Kernel 5: finalize.
// self_mean[i] = deg[i]^2 * h[i] / max(deg,1) == deg[i]*h[i]  (exact, incl. deg==0)
// out = (agg + 0.5*deg*h - bn_mean) * (gamma * rsqrt(var+eps)) + beta
// ---------------------------------------------------------------------------
__global__ __launch_bounds__(256) void finalize_kernel(const float* __restrict__ agg,
                                                       const float* __restrict__ h,
                                                       const float* __restrict__ deg,
                                                       const float* __restrict__ gamma,
                                                       const float* __restrict__ beta,
                                                       const float* __restrict__ mean,
                                                       const float* __restrict__ var,
                                                       float* __restrict__ out,
                                                       long long total) {
  long long i = (long long)blockIdx.x * blockDim.x + threadIdx.x;
  if (i >= total) return;
  int node = (int)(i >> 6);
  int j = (int)(i & 63);
  float a = agg[i] + 0.5f * deg[node] * h[i];
  out[i] = (a - mean[j]) * (gamma[j] * rsqrtf(var[j] + BN_EPS)) + beta[j];
}

// ---------------------------------------------------------------------------
extern "C" void kernel_launch(void* const* d_in, const int* in_sizes, int n_in,
                              void* d_out, int out_size, void* d_ws, size_t ws_size,
                              hipStream_t stream) {
  const float* X     = (const float*)d_in[0];   // node_features (N,128)
  const float* W     = (const float*)d_in[1];   // (128,64)
  const float* b     = (const float*)d_in[2];   // (64,)
  const float* gamma = (const float*)d_in[3];
  const float* beta  = (const float*)d_in[4];
  const float* mean  = (const float*)d_in[5];
  const float* var   = (const float*)d_in[6];
  const int*   ei    = (const int*)d_in[7];     // (E,2) [src,dst]

  const int F  = 128, Hc = 64;
  const int N  = in_sizes[0] / F;
  const int E  = in_sizes[7] / 2;

  float* ws   = (float*)d_ws;
  float* h    = ws;                       // N*64
  float* agg  = h   + (size_t)N * Hc;     // N*64
  float* deg  = agg + (size_t)N * Hc;     // N
  float* invd = deg + N;                  // N

  // zero agg + deg (contiguous region); capturable memset node
  hipMemsetAsync(agg, 0, ((size_t)N * Hc + (size_t)N) * sizeof(float), stream);

  deg_kernel<<<(E + 255) / 256, 256, 0, stream>>>(ei, deg, E);
  invd_kernel<<<(N + 255) / 256, 256, 0, stream>>>(deg, invd, N);

  const int ntiles = ((N + 15) / 16) * 4;
  gemm_wmma<<<(ntiles + 7) / 8, 256, 0, stream>>>(X, W, b, h, N);

  const long long ethreads = (long long)E * 32;
  edge_scatter<<<(int)((ethreads + 255) / 256), 256, 0, stream>>>(ei, h, invd, agg, E);

  const long long total = (long long)N * Hc;
  finalize_kernel<<<(int)((total + 255) / 256), 256, 0, stream>>>(
      agg, h, deg, gamma, beta, mean, var, (float*)d_out, total);
}